// ExponentialTrajectoryFit_59665685676057
// MI455X (gfx1250) — compile-verified
//
#include <hip/hip_runtime.h>
#include <hip/hip_bf16.h>

typedef __attribute__((ext_vector_type(2))) float v2f;
typedef __attribute__((ext_vector_type(4))) float v4f;
typedef __attribute__((ext_vector_type(8))) float v8f;

#define DIM 8192
#define NTHREADS 1024
#define NWAVES (NTHREADS / 32)
#define KCHUNKS ((DIM / 4) / NWAVES)   // 64 iterations per wave, compile-time

// Single fused kernel (problem is launch-latency bound: ~192 KB total traffic
// = ~8 ns at 23.3 TB/s, so one launch, one workgroup):
//   Phase 1: 32 waves accumulate the 5x5 Gram of S = {d0,d1,d2,x0,x1} with
//            V_WMMA_F32_16X16X4_F32. Per-lane source selection hoisted out of
//            the loop (row value = sp*P[k] - sq*Q[k], loop-invariant p/q/sp/sq),
//            constant trip count + constant stride so unrolling and immediate-
//            offset addressing actually fire.
//   Phase 2: 10-thread parallel cross-wave reduce, thread 0 does the 2x2
//            Woodbury solve -> alpha, beta.
//   Phase 3: out = -d0 + alpha*d1 + beta*d2, vectorized b128.
__global__ __launch_bounds__(NTHREADS)
void ExponentialTrajectoryFit_fused(const float* __restrict__ x0,
                                    const float* __restrict__ x1,
                                    const float* __restrict__ x2,
                                    const float* __restrict__ x3,
                                    float* __restrict__ out) {
    const int tid  = threadIdx.x;
    const int wave = tid >> 5;
    const int lane = tid & 31;
    const int m    = lane & 15;          // row id of A / col id of B
    const int kh   = (lane >> 4) << 1;   // lanes 0-15 -> K{0,1}, lanes 16-31 -> K{2,3}

    // ---- loop-invariant per-lane source selection: a = sp*P - sq*Q ----
    const float* p;
    const float* q;
    float sp, sq;
    if (m == 0)      { p = x1; q = x0; sp = 1.0f; sq = 1.0f; }   // d0
    else if (m == 1) { p = x2; q = x1; sp = 1.0f; sq = 1.0f; }   // d1
    else if (m == 2) { p = x3; q = x2; sp = 1.0f; sq = 1.0f; }   // d2
    else if (m == 3) { p = x0; q = x0; sp = 1.0f; sq = 0.0f; }   // x0
    else if (m == 4) { p = x1; q = x0; sp = 1.0f; sq = 0.0f; }   // x1
    else             { p = x0; q = x0; sp = 0.0f; sq = 0.0f; }   // zero pad

    // Per-lane base; successive iterations advance by a constant 128 floats
    // (512 B) -> foldable into the load immediate offset.
    const float* pp = p + wave * 4 + kh;
    const float* qq = q + wave * 4 + kh;

    // ---------------- Phase 1: Gram via WMMA f32 16x16x4 ----------------
    v8f acc = {};
    #pragma unroll 8
    for (int it = 0; it < KCHUNKS; ++it) {
        const v2f P = *(const v2f*)(pp + it * (4 * NWAVES));   // 8B aligned
        const v2f Q = *(const v2f*)(qq + it * (4 * NWAVES));
        v2f a;
        a[0] = sp * P[0] - sq * Q[0];
        a[1] = sp * P[1] - sq * Q[1];
        // Symmetric product: A (16x4) and B (4x16) lane layouts coincide.
        acc = __builtin_amdgcn_wmma_f32_16x16x4_f32(
            /*neg_a=*/false, a, /*neg_b=*/false, a,
            /*c_mod=*/(short)0, acc, /*reuse_a=*/false, /*reuse_b=*/false);
    }

    // Per-wave partials: rows 0,1 x cols 0..4 of the Gram (10 floats/wave).
    __shared__ float partial[NWAVES][10];
    __shared__ float gsum[10];
    __shared__ float coeff[2];
    if (lane < 5) {
        partial[wave][lane * 2 + 0] = acc[0];   // G[0][lane]
        partial[wave][lane * 2 + 1] = acc[1];   // G[1][lane]
    }
    __syncthreads();

    // ---------------- Phase 2: parallel reduce + Woodbury solve ----------------
    if (tid < 10) {
        float s = 0.0f;
        #pragma unroll
        for (int w = 0; w < NWAVES; ++w) s += partial[w][tid];
        gsum[tid] = s;
    }
    __syncthreads();
    if (tid == 0) {
        // gsum[2n+r] = (row r of {d0,d1}) . (col n of {d0,d1,d2,x0,x1})
        const float g00 = gsum[0], g10 = gsum[1];   // d0.d0, d1.d0
        const float g01 = gsum[2], g11 = gsum[3];   // d0.d1, d1.d1
        const float q02 = gsum[4], q12 = gsum[5];   // d0.d2, d1.d2
        const float p00 = gsum[6], p01 = gsum[7];   // d0.x0, d1.x0
        const float p10 = gsum[8], p11 = gsum[9];   // d0.x1, d1.x1

        const float det = g00 * g11 - g01 * g10;
        const float id  = 1.0f / det;
        // a = G^-1 * (D1^T x0)
        const float a0 = id * ( g11 * p00 - g01 * p01);
        const float a1 = id * (-g10 * p00 + g00 * p01);
        // r = D1^T z = D1^T x1 - T*a, T = D1^T U = [[g01,q02],[g11,q12]]
        const float r0 = p10 - (g01 * a0 + q02 * a1);
        const float r1 = p11 - (g11 * a0 + q12 * a1);
        // v = V^T z = G^-1 * r
        const float v0 = id * ( g11 * r0 - g01 * r1);
        const float v1 = id * (-g10 * r0 + g00 * r1);
        // S = V^T U = G^-1 * T ; W = I - S
        const float s00 = id * ( g11 * g01 - g01 * g11);
        const float s10 = id * (-g10 * g01 + g00 * g11);
        const float s01 = id * ( g11 * q02 - g01 * q12);
        const float s11 = id * (-g10 * q02 + g00 * q12);
        const float w00 = 1.0f - s00, w01 = -s01;
        const float w10 = -s10,       w11 = 1.0f - s11;
        const float iw  = 1.0f / (w00 * w11 - w01 * w10);
        // c = W^-1 * v
        const float c0 = iw * ( w11 * v0 - w01 * v1);
        const float c1 = iw * (-w10 * v0 + w00 * v1);

        coeff[0] = a0 - c0;   // alpha (d1 coefficient)
        coeff[1] = a1 - c1;   // beta  (d2 coefficient)
    }
    __syncthreads();

    // -------- Phase 3: out = -d0 + alpha*d1 + beta*d2 (b128 vectorized) --------
    const float alpha = coeff[0];
    const float beta  = coeff[1];
    const int base = tid * 8;          // 8 elements per thread, DIM = 8*NTHREADS
    #pragma unroll
    for (int j = 0; j < 2; ++j) {
        const int i = base + j * 4;
        const v4f e0 = *(const v4f*)(x0 + i);
        const v4f e1 = *(const v4f*)(x1 + i);
        const v4f e2 = *(const v4f*)(x2 + i);
        const v4f e3 = *(const v4f*)(x3 + i);
        const v4f r  = (e0 - e1) + alpha * (e2 - e1) + beta * (e3 - e2);
        *(v4f*)(out + i) = r;
    }
}

extern "C" void kernel_launch(void* const* d_in, const int* in_sizes, int n_in,
                              void* d_out, int out_size, void* d_ws, size_t ws_size,
                              hipStream_t stream) {
    (void)in_sizes; (void)n_in; (void)out_size; (void)d_ws; (void)ws_size;
    const float* x0 = (const float*)d_in[0];
    const float* x1 = (const float*)d_in[1];
    const float* x2 = (const float*)d_in[2];
    const float* x3 = (const float*)d_in[3];
    float* out = (float*)d_out;
    ExponentialTrajectoryFit_fused<<<1, NTHREADS, 0, stream>>>(x0, x1, x2, x3, out);
}